// ExpertChoice_73366631350526
// MI455X (gfx1250) — compile-verified
//
#include <hip/hip_runtime.h>
#include <math.h>

// Problem constants (from reference)
#define B_   64
#define N_   32
#define D_   384
#define E_   8
#define K_   8
#define C_   1000
#define KD_  3072   // K*D
#define ND_  12288  // N*D

typedef float v2f __attribute__((ext_vector_type(2)));
typedef float v8f __attribute__((ext_vector_type(8)));

__device__ __forceinline__ float gelu_exact(float x) {
    // exact gelu: 0.5*x*(1+erf(x/sqrt(2)))
    return 0.5f * x * (1.0f + erff(x * 0.70710678118654752440f));
}

// ---------------------------------------------------------------------------
// 1) Routing: logits[b][e][n] = x[b,n,:]·emb[e,:]; top-8 indices per (b,e).
//    Softmax is monotonic -> top-k(softmax) == top-k(logits); G is unused.
//    One block per b, 256 threads = 8 waves; wave e owns expert e, lane = n.
// ---------------------------------------------------------------------------
__global__ void route_topk_kernel(const float* __restrict__ x,
                                  const float* __restrict__ emb,
                                  int* __restrict__ topk_idx) {
    __shared__ float logits[E_][N_];
    const int b = blockIdx.x;
    const int t = threadIdx.x;
    const int e = t >> 5;
    const int n = t & 31;

    const float4* xr4 = (const float4*)(x + ((size_t)b * N_ + n) * D_);
    const float4* er4 = (const float4*)(emb + (size_t)e * D_);
    float s = 0.f;
    #pragma unroll 4
    for (int d = 0; d < D_ / 4; ++d) {
        float4 a = xr4[d], w = er4[d];
        s += a.x * w.x + a.y * w.y + a.z * w.z + a.w * w.w;
    }
    logits[e][n] = s;
    __syncthreads();

    // iterative argmax top-8 within the wave (lane n holds logit n)
    float v = logits[e][n];
    const int lane = n;
    for (int k = 0; k < K_; ++k) {
        float bv = v;
        int   bi = lane;
        for (int off = 16; off > 0; off >>= 1) {
            float ov = __shfl_xor(bv, off, 32);
            int   oi = __shfl_xor(bi, off, 32);
            // strict max; ties pick the lower index (matches jax top_k)
            if (ov > bv || (ov == bv && oi < bi)) { bv = ov; bi = oi; }
        }
        if (lane == 0) topk_idx[((size_t)b * E_ + e) * K_ + k] = bi;
        if (lane == bi) v = -INFINITY;
    }
}

// ---------------------------------------------------------------------------
// 2) Gather selected tokens: sel[e][b][k*D..] = x[b, I[b,e,k], :]
//    One block per (b,e,k), 96 threads copy 384 floats as float4.
// ---------------------------------------------------------------------------
__global__ void gather_kernel(const float* __restrict__ x,
                              const int* __restrict__ topk_idx,
                              float* __restrict__ sel) {
    const int flat = blockIdx.x;
    const int k = flat % K_;
    const int e = (flat / K_) % E_;
    const int b = flat / (K_ * E_);
    const int tok = topk_idx[((size_t)b * E_ + e) * K_ + k];

    const float4* src = (const float4*)(x + ((size_t)b * N_ + tok) * D_);
    float4* dst = (float4*)(sel + ((size_t)e * B_ + b) * KD_ + (size_t)k * D_);
    const int t = threadIdx.x;
    if (t < D_ / 4) dst[t] = src[t];
}

// ---------------------------------------------------------------------------
// 3) Batched GEMM + bias + optional exact-GELU, fp32 via V_WMMA_F32_16X16X4_F32.
//    C[e] (64 x Ndim) = act(A[e] (64 x Kdim) @ W[e] (Kdim x Ndim) + bias[e])
//    Block = 256 threads (8 waves); block owns all 64 rows x 128-col panel so
//    each weight element streams from HBM exactly once; A stays in L2.
//    All loads are SGPR-base + u32 lane-offset (scale_offset form): W offsets
//    in float units (b32 loads), A offsets in v2f units (b64 loads).
//    Inner loop: 4x b64 + 2x b32 loads, 4x wmma, 2x u32 adds.
//    GUARD=false instantiation (Ndim % 128 == 0) has no bounds code at all.
// ---------------------------------------------------------------------------
template <bool GUARD>
__global__ void gemm_wmma_kernel(const float* __restrict__ A,
                                 const float* __restrict__ W,
                                 const float* __restrict__ bias,
                                 float* __restrict__ Cm,
                                 int Kdim, int Ndim, int applyGelu,
                                 long long strideA, long long strideW,
                                 long long strideBias, long long strideC) {
    const int e = blockIdx.y;
    const float* __restrict__ Ab = A + (long long)e * strideA;
    const float* __restrict__ Wb = W + (long long)e * strideW;
    const float* __restrict__ Bb = bias + (long long)e * strideBias;
    float* __restrict__ Cb = Cm + (long long)e * strideC;

    const int wave = threadIdx.x >> 5;
    const int lane = threadIdx.x & 31;
    const int n0   = blockIdx.x * 128 + wave * 16;
    const int col  = n0 + (lane & 15);
    const bool colok = GUARD ? (col < Ndim) : true;
    const int  colc  = GUARD ? (colok ? col : (Ndim - 1)) : col;  // clamped, always legal
    const int half = lane >> 4;        // 0: K pair {0,1}, 1: K pair {2,3}
    const int arow = lane & 15;

    // Uniform (SGPR) base pointers; per-lane state is just two u32 offsets.
    const v2f* __restrict__ A0 = (const v2f*)(Ab);
    const v2f* __restrict__ A1 = (const v2f*)(Ab + (size_t)16 * Kdim);
    const v2f* __restrict__ A2 = (const v2f*)(Ab + (size_t)32 * Kdim);
    const v2f* __restrict__ A3 = (const v2f*)(Ab + (size_t)48 * Kdim);
    const float* __restrict__ Wb1 = Wb + Ndim;   // row kb+1

    // A offset in v2f (8-byte) units: row stride = Kdim/2, step = 2 per K-iter.
    unsigned aoff = (unsigned)arow * (unsigned)(Kdim >> 1) + (unsigned)half;
    // W offset in float (4-byte) units.
    unsigned woff = (unsigned)(half << 1) * (unsigned)Ndim + (unsigned)colc;
    const unsigned wstep = 4u * (unsigned)Ndim;

    v8f acc0 = {}, acc1 = {}, acc2 = {}, acc3 = {};

    #pragma unroll 2
    for (int kk = 0; kk < Kdim; kk += 4) {
        const float bx = Wb[woff];
        const float by = Wb1[woff];
        v2f bfrag;
        bfrag.x = GUARD ? (colok ? bx : 0.f) : bx;   // v_cndmask, no branch
        bfrag.y = GUARD ? (colok ? by : 0.f) : by;
        const v2f a0 = A0[aoff];
        const v2f a1 = A1[aoff];
        const v2f a2 = A2[aoff];
        const v2f a3 = A3[aoff];

        acc0 = __builtin_amdgcn_wmma_f32_16x16x4_f32(false, a0, false, bfrag, (short)0, acc0, false, false);
        acc1 = __builtin_amdgcn_wmma_f32_16x16x4_f32(false, a1, false, bfrag, (short)0, acc1, false, false);
        acc2 = __builtin_amdgcn_wmma_f32_16x16x4_f32(false, a2, false, bfrag, (short)0, acc2, false, false);
        acc3 = __builtin_amdgcn_wmma_f32_16x16x4_f32(false, a3, false, bfrag, (short)0, acc3, false, false);

        aoff += 2u;
        woff += wstep;
    }

    if (!GUARD || colok) {
        const float bv = Bb[col];
        const int rofs = half << 3;  // lanes 16-31 hold rows +8
        #pragma unroll
        for (int m = 0; m < 4; ++m) {
            v8f a = (m == 0) ? acc0 : (m == 1) ? acc1 : (m == 2) ? acc2 : acc3;
            unsigned ooff = (unsigned)(m * 16 + rofs) * (unsigned)Ndim + (unsigned)col;
            #pragma unroll
            for (int r = 0; r < 8; ++r) {
                float vv = a[r] + bv;
                if (applyGelu) vv = gelu_exact(vv);
                Cb[ooff] = vv;
                ooff += (unsigned)Ndim;
            }
        }
    }
}

// ---------------------------------------------------------------------------
// 4) Gate second layer + softmax: w[b][e] = softmax_e(g1[b,:]·sw_w2[:,e]+b2[e])
//    One block per b; wave e computes one dot of length 12288.
// ---------------------------------------------------------------------------
__global__ void gate2_softmax_kernel(const float* __restrict__ g1,
                                     const float* __restrict__ sw_w2,
                                     const float* __restrict__ sw_b2,
                                     float* __restrict__ wgt) {
    __shared__ float lg[E_];
    const int b = blockIdx.x;
    const int e = threadIdx.x >> 5;
    const int lane = threadIdx.x & 31;

    const float* g = g1 + (size_t)b * ND_;
    float acc = 0.f;
    for (int i = lane; i < ND_; i += 32)
        acc += g[i] * sw_w2[(size_t)i * E_ + e];
    for (int off = 16; off > 0; off >>= 1)
        acc += __shfl_xor(acc, off, 32);
    if (lane == 0) lg[e] = acc + sw_b2[e];
    __syncthreads();

    if (threadIdx.x == 0) {
        float mx = lg[0];
        for (int i = 1; i < E_; ++i) mx = fmaxf(mx, lg[i]);
        float ex[E_];
        float sum = 0.f;
        for (int i = 0; i < E_; ++i) { ex[i] = expf(lg[i] - mx); sum += ex[i]; }
        const float inv = 1.f / sum;
        for (int i = 0; i < E_; ++i) wgt[(size_t)b * E_ + i] = ex[i] * inv;
    }
}

// ---------------------------------------------------------------------------
// 5) Expert-weighted sum: ws[b][k] = sum_e er[e][b][k] * w[b][e]
// ---------------------------------------------------------------------------
__global__ void wsum_kernel(const float* __restrict__ er,
                            const float* __restrict__ wgt,
                            float* __restrict__ wsbuf) {
    const unsigned gid = blockIdx.x * blockDim.x + threadIdx.x;
    if (gid >= (unsigned)(B_ * KD_)) return;
    const unsigned b = gid / KD_;
    float s = 0.f;
    #pragma unroll
    for (int e = 0; e < E_; ++e)
        s += er[(unsigned)e * (B_ * KD_) + gid] * wgt[b * E_ + e];
    wsbuf[gid] = s;
}

// ---------------------------------------------------------------------------
// Launch
// ---------------------------------------------------------------------------
extern "C" void kernel_launch(void* const* d_in, const int* in_sizes, int n_in,
                              void* d_out, int out_size, void* d_ws, size_t ws_size,
                              hipStream_t stream) {
    const float* x     = (const float*)d_in[0];
    const float* emb   = (const float*)d_in[1];
    const float* w1    = (const float*)d_in[2];
    const float* b1    = (const float*)d_in[3];
    const float* w2    = (const float*)d_in[4];
    const float* b2    = (const float*)d_in[5];
    const float* sw_w1 = (const float*)d_in[6];
    const float* sw_b1 = (const float*)d_in[7];
    const float* sw_w2 = (const float*)d_in[8];
    const float* sw_b2 = (const float*)d_in[9];
    const float* ch_w1 = (const float*)d_in[10];
    const float* ch_b1 = (const float*)d_in[11];
    const float* ch_w2 = (const float*)d_in[12];
    const float* ch_b2 = (const float*)d_in[13];
    float* out = (float*)d_out;

    // Workspace carve-up (total ~23.5 MB)
    char* ws = (char*)d_ws;
    size_t off = 0;
    auto carve = [&](size_t bytes) -> void* {
        void* p = ws + off;
        off += (bytes + 255) & ~(size_t)255;
        return p;
    };
    int*   idx   = (int*)  carve((size_t)B_ * E_ * K_ * sizeof(int));
    float* sel   = (float*)carve((size_t)E_ * B_ * KD_ * sizeof(float));
    float* h     = (float*)carve((size_t)E_ * B_ * KD_ * sizeof(float));
    float* er    = (float*)carve((size_t)E_ * B_ * KD_ * sizeof(float));
    float* g1    = (float*)carve((size_t)B_ * ND_ * sizeof(float));
    float* wgt   = (float*)carve((size_t)B_ * E_ * sizeof(float));
    float* wsbuf = (float*)carve((size_t)B_ * KD_ * sizeof(float));
    float* t1    = (float*)carve((size_t)B_ * KD_ * sizeof(float));
    (void)ws_size; (void)in_sizes; (void)n_in; (void)out_size;

    // 1) routing + top-k indices
    route_topk_kernel<<<B_, 256, 0, stream>>>(x, emb, idx);
    // 2) gather selected tokens -> sel (E,B,KD)
    gather_kernel<<<B_ * E_ * K_, 96, 0, stream>>>(x, idx, sel);
    // 3a) expert layer 1: h = gelu(sel @ w1 + b1), batched over E
    gemm_wmma_kernel<false><<<dim3(KD_ / 128, E_), 256, 0, stream>>>(
        sel, w1, b1, h, KD_, KD_, 1,
        (long long)B_ * KD_, (long long)KD_ * KD_, (long long)KD_, (long long)B_ * KD_);
    // 3b) expert layer 2: er = h @ w2 + b2 (kept in (E,B,KD))
    gemm_wmma_kernel<false><<<dim3(KD_ / 128, E_), 256, 0, stream>>>(
        h, w2, b2, er, KD_, KD_, 0,
        (long long)B_ * KD_, (long long)KD_ * KD_, (long long)KD_, (long long)B_ * KD_);
    // 4a) gate layer 1: g1 = gelu(xf @ sw_w1 + sw_b1)  (xf = x viewed (64,12288))
    gemm_wmma_kernel<false><<<dim3(ND_ / 128, 1), 256, 0, stream>>>(
        x, sw_w1, sw_b1, g1, ND_, ND_, 1, 0, 0, 0, 0);
    // 4b) gate layer 2 + softmax -> wgt (B,E)
    gate2_softmax_kernel<<<B_, 256, 0, stream>>>(g1, sw_w2, sw_b2, wgt);
    // 5) expert-weighted sum -> wsbuf (B,KD)
    wsum_kernel<<<(B_ * KD_ + 255) / 256, 256, 0, stream>>>(er, wgt, wsbuf);
    // 6a) head layer 1: t1 = gelu(wsbuf @ ch_w1 + ch_b1)
    gemm_wmma_kernel<false><<<dim3(KD_ / 128, 1), 256, 0, stream>>>(
        wsbuf, ch_w1, ch_b1, t1, KD_, KD_, 1, 0, 0, 0, 0);
    // 6b) head layer 2: out = t1 @ ch_w2 + ch_b2  (N=1000, guarded path)
    gemm_wmma_kernel<true><<<dim3((C_ + 127) / 128, 1), 256, 0, stream>>>(
        t1, ch_w2, ch_b2, out, KD_, C_, 0, 0, 0, 0, 0);
}